// GatedGraphX_17205638988469
// MI455X (gfx1250) — compile-verified
//
#include <hip/hip_runtime.h>
#include <math.h>

#define NN 2
#define TT 4096
#define II 64
#define HH 64
#define EE 12
#define SS 31
#define LL 256
#define ITERS 10
#define NT (NN * TT)
#define K1 (EE * HH)   /* 768 */
#define C3 (3 * HH)    /* 192 */
#define GWAVES 4
#define LPAD 202       /* 192+10: lanes 0-15 and 16-31 hit disjoint LDS banks */

typedef float v2f __attribute__((ext_vector_type(2)));
typedef float v8f __attribute__((ext_vector_type(8)));

__device__ __forceinline__ v8f wmma_f32(v2f a, v2f b, v8f c) {
  // D = A(16x4) * B(4x16) + C, full fp32
  return __builtin_amdgcn_wmma_f32_16x16x4_f32(false, a, false, b, (short)0, c,
                                               false, false);
}

__device__ __forceinline__ float sigmoidf_(float x) {
  return 1.0f / (1.0f + expf(-x));
}

// ---------------------------------------------------------------------------
// Kernel 0: padded-row mask (input rows that are all zero), computed once.
// ---------------------------------------------------------------------------
__global__ void mask_kernel(const float* __restrict__ input,
                            unsigned char* __restrict__ mask) {
  int row = blockIdx.x * blockDim.x + threadIdx.x;
  if (row >= NT) return;
  const float* p = input + (size_t)row * II;
  int padded = 1;
  for (int i = 0; i < II; ++i) padded &= (p[i] == 0.0f);
  mask[row] = (unsigned char)padded;
}

// ---------------------------------------------------------------------------
// Kernel 1: act[n, t, e*64+h] = sum_i edge[n,s(t),e,i,p(t)] * h[n,128*s(t)+i,h]
//           + ba[e,h]
// Only combine-selected output rows j are computed:
//   s=0: j in [0,192); middle: j in [64,192); s=S-1: j in [64,256); t = 128s+j
// One block per (s,e,n); 8 waves. Each wave fixes nt = wave&3 so its B
// fragment is shared across all its M tiles: TPW accumulators, one K loop,
// 2 B loads + 2*TPW A loads per 4-wide K step (~2.3-2.5 dwords per WMMA).
// ---------------------------------------------------------------------------
template <int TPW>
__device__ __forceinline__ void edge_compute(
    const float* __restrict__ eb,   // edge matrix base (L x L) for (n,s,e)
    const float* __restrict__ hb,   // h slice base (256 x 64)
    const float* __restrict__ ba, float* __restrict__ act, int s, int e, int n,
    int row_lo, int wave, int lhalf, int lmod) {
  const int nt = wave & 3;
  const int c = nt * 16 + lmod;                     // output col h
  const int mbase = row_lo + ((wave >> 2) * 16) + lmod;  // + j*32 per tile

  v8f acc[TPW];
#pragma unroll
  for (int j = 0; j < TPW; ++j) acc[j] = (v8f){};

  const float* hb0 = hb + (size_t)(2 * lhalf) * HH + c;
  const float* ea0 = eb + (size_t)(2 * lhalf) * LL + mbase;

#pragma unroll 2
  for (int k = 0; k < LL; k += 4) {
    v2f b;
    b.x = hb0[(size_t)k * HH];
    b.y = hb0[(size_t)(k + 1) * HH];
#pragma unroll
    for (int j = 0; j < TPW; ++j) {
      v2f a;
      a.x = ea0[(size_t)k * LL + j * 32];
      a.y = ea0[(size_t)(k + 1) * LL + j * 32];
      acc[j] = wmma_f32(a, b, acc[j]);
    }
  }

  const float bias = ba[e * HH + c];
#pragma unroll
  for (int j = 0; j < TPW; ++j) {
    const int mt = (wave >> 2) + 2 * j;
    const int t0 = s * 128 + row_lo + mt * 16 + 8 * lhalf;
    float* ob = act + ((size_t)(n * TT + t0)) * K1 + (size_t)e * HH + c;
#pragma unroll
    for (int r = 0; r < 8; ++r) ob[(size_t)r * K1] = acc[j][r] + bias;
  }
}

__global__ __launch_bounds__(256) void edge_kernel(
    const float* __restrict__ edge,   // (N,S,E,L,L)
    const float* __restrict__ h,      // (N,T,H)
    const float* __restrict__ ba,     // (E,H)
    float* __restrict__ act) {        // (N*T, 768)
  const int s = blockIdx.x, e = blockIdx.y, n = blockIdx.z;
  const int wave = threadIdx.x >> 5;
  const int lane = threadIdx.x & 31;
  const int lhalf = lane >> 4, lmod = lane & 15;

  const float* eb = edge + (((size_t)n * SS + s) * EE + e) * (size_t)(LL * LL);
  const float* hb = h + ((size_t)n * TT + (size_t)s * 128) * HH;

  if (s == 0 || s == SS - 1) {
    // 192 selected rows -> 12 M tiles -> 6 per wave-pair slot
    edge_compute<6>(eb, hb, ba, act, s, e, n, (s == 0) ? 0 : 64, wave, lhalf,
                    lmod);
  } else {
    // 128 selected rows -> 8 M tiles -> 4 per wave-pair slot
    edge_compute<4>(eb, hb, ba, act, s, e, n, 64, wave, lhalf, lmod);
  }
}

// ---------------------------------------------------------------------------
// Kernel 2: fused GRU update for 16-row strips.
//  g1 = [act | input] @ [W ; input_wzrh]   (K=768+64, 192 cols) -> LDS
//  g2 = h @ [uz_ur | uh]                   (K=64,     192 cols) -> regs
//  z = sig(g1[:64]+g2[:64]); r = sig(g1[64:128]+g2[64:128])
//  h~ = tanh(g1[128:]+r*g2[128:]);  h' = (1-z)h + z h~ ; padded -> 0
// ---------------------------------------------------------------------------
__global__ __launch_bounds__(GWAVES * 32) void gru_kernel(
    const float* __restrict__ act,    // (NT, 768)
    const float* __restrict__ W,      // (768, 192)
    const float* __restrict__ h,      // (NT, 64)
    const float* __restrict__ uzr,    // (64, 128)
    const float* __restrict__ uh,     // (64, 64)
    const float* __restrict__ input,  // (NT, 64)
    const float* __restrict__ iw,     // (64, 192)
    const unsigned char* __restrict__ mask,
    float* __restrict__ hout) {       // (NT, 64)
  __shared__ float lds[GWAVES][16][LPAD];
  const int wave = threadIdx.x >> 5;
  const int lane = threadIdx.x & 31;
  const int lhalf = lane >> 4, lmod = lane & 15;
  const int row0 = (blockIdx.x * GWAVES + wave) * 16;

  v8f acc[12];
#pragma unroll
  for (int ct = 0; ct < 12; ++ct) acc[ct] = (v8f){};

  // ---- g1: act @ W ----
  const float* arow = act + (size_t)(row0 + lmod) * K1 + 2 * lhalf;
#pragma unroll 2
  for (int k = 0; k < K1; k += 4) {
    v2f a = *(const v2f*)(arow + k);
    const float* w0 = W + (size_t)(k + 2 * lhalf) * C3 + lmod;
#pragma unroll
    for (int ct = 0; ct < 12; ++ct) {
      v2f b;
      b.x = w0[ct * 16];
      b.y = w0[C3 + ct * 16];
      acc[ct] = wmma_f32(a, b, acc[ct]);
    }
  }
  // ---- g1 += input @ input_wzrh (folded inp_proj) ----
  const float* xrow = input + (size_t)(row0 + lmod) * II + 2 * lhalf;
#pragma unroll 2
  for (int k = 0; k < II; k += 4) {
    v2f a = *(const v2f*)(xrow + k);
    const float* w0 = iw + (size_t)(k + 2 * lhalf) * C3 + lmod;
#pragma unroll
    for (int ct = 0; ct < 12; ++ct) {
      v2f b;
      b.x = w0[ct * 16];
      b.y = w0[C3 + ct * 16];
      acc[ct] = wmma_f32(a, b, acc[ct]);
    }
  }
  // stash g1 in LDS (each wave reads only its own rows; no barrier needed)
#pragma unroll
  for (int ct = 0; ct < 12; ++ct) {
#pragma unroll
    for (int r = 0; r < 8; ++r)
      lds[wave][r + 8 * lhalf][ct * 16 + lmod] = acc[ct][r];
  }

  // ---- g2: h @ [uz_ur | uh] ----
#pragma unroll
  for (int ct = 0; ct < 12; ++ct) acc[ct] = (v8f){};
  const float* hrow = h + (size_t)(row0 + lmod) * HH + 2 * lhalf;
#pragma unroll 2
  for (int k = 0; k < HH; k += 4) {
    v2f a = *(const v2f*)(hrow + k);
    const float* u0 = uzr + (size_t)(k + 2 * lhalf) * (2 * HH) + lmod;
#pragma unroll
    for (int ct = 0; ct < 8; ++ct) {
      v2f b;
      b.x = u0[ct * 16];
      b.y = u0[2 * HH + ct * 16];
      acc[ct] = wmma_f32(a, b, acc[ct]);
    }
    const float* u1 = uh + (size_t)(k + 2 * lhalf) * HH + lmod;
#pragma unroll
    for (int ct = 8; ct < 12; ++ct) {
      v2f b;
      b.x = u1[(ct - 8) * 16];
      b.y = u1[HH + (ct - 8) * 16];
      acc[ct] = wmma_f32(a, b, acc[ct]);
    }
  }

  // ---- gate epilogue ----
#pragma unroll
  for (int ct = 0; ct < 4; ++ct) {
    const int k = ct * 16 + lmod;
#pragma unroll
    for (int r = 0; r < 8; ++r) {
      const int lrow = r + 8 * lhalf;
      const int row = row0 + lrow;
      const float az = lds[wave][lrow][k];
      const float ar = lds[wave][lrow][64 + k];
      const float ah = lds[wave][lrow][128 + k];
      const float z = sigmoidf_(az + acc[ct][r]);
      const float rr = sigmoidf_(ar + acc[ct + 4][r]);
      const float ht = tanhf(ah + rr * acc[ct + 8][r]);
      const float hv = h[(size_t)row * HH + k];
      float hn = (1.0f - z) * hv + z * ht;
      if (mask[row]) hn = 0.0f;
      hout[(size_t)row * HH + k] = hn;
    }
  }
}

// ---------------------------------------------------------------------------
extern "C" void kernel_launch(void* const* d_in, const int* in_sizes, int n_in,
                              void* d_out, int out_size, void* d_ws,
                              size_t ws_size, hipStream_t stream) {
  (void)in_sizes; (void)n_in; (void)out_size; (void)ws_size;
  const float* input  = (const float*)d_in[0];
  const float* hidden = (const float*)d_in[1];
  const float* edge   = (const float*)d_in[2];
  const float* ba     = (const float*)d_in[3];
  const float* W      = (const float*)d_in[4];  // (E,H,3H) == (768,192)
  const float* uzr    = (const float*)d_in[5];
  const float* uh     = (const float*)d_in[6];
  const float* iw     = (const float*)d_in[7];
  float* out = (float*)d_out;

  float* act = (float*)d_ws;                       // NT * 768 floats
  float* hb0 = act + (size_t)NT * K1;              // NT * 64
  float* hb1 = hb0 + (size_t)NT * HH;              // NT * 64
  unsigned char* mask = (unsigned char*)(hb1 + (size_t)NT * HH);

  mask_kernel<<<NT / 256, 256, 0, stream>>>(input, mask);

  const float* hcur = hidden;
  for (int it = 0; it < ITERS; ++it) {
    edge_kernel<<<dim3(SS, EE, NN), 256, 0, stream>>>(edge, hcur, ba, act);
    float* hnext = (it == ITERS - 1) ? out : ((it & 1) ? hb1 : hb0);
    gru_kernel<<<NT / (GWAVES * 16), GWAVES * 32, 0, stream>>>(
        act, W, hcur, uzr, uh, input, iw, mask, hnext);
    hcur = hnext;
  }
}